// Model_89069031784536
// MI455X (gfx1250) — compile-verified
//
#include <hip/hip_runtime.h>
#include <hip/hip_bf16.h>

// Shapes: B=128, S=720, C=862, N=B*C=110336, WINDOW=12, RESERVOIR=25,
// T=60 steps, WASHOUT=10 -> 50 kept states, pred_seg=60.
//
// Pipeline:
//  K1: ws[0..3328)   = Gpad[64][52]  (G[p,t]=fc2_w@fc1_w, col50=1 (mean lane), col51=0, pads=0)
//      ws[3328..4096)= outbias[64][12] (proj_b[w] + rowsum(proj_w[w,:]) * (fc2_b[p] + fc2_w@fc1_b))
//      (d_ws requirement: 16 KiB)
//  K2: per-thread n: mean, ESN recurrence (weights in LDS, v_tanh_f32),
//      q_t[w] = proj_w @ h_t staged into d_out rows t'*12+w (t'<50),
//      mean staged into d_out rows 600..611 (t'=50).
//  K3: per-wave WMMA GEMM D[p,n] = Gpad[p,t] x Qt[t,n] (f32 16x16x4),
//      K=52 (50 q cols + mean col + zero col), final = D + outbias, overwrite d_out.
//      Each (n,w) slice of d_out is read (q) and written (out) by exactly one wave -> race-free.

typedef __attribute__((ext_vector_type(2))) float v2f;
typedef __attribute__((ext_vector_type(8))) float v8f;

#define NB 128
#define NC 862
#define NSEQ 110336          // NB*NC
#define SROW 620640          // 720*862 floats per batch image
#define PSTEP 10344          // 12*862 floats between p (or t') rows for fixed w

__device__ __forceinline__ float tanh_hw(float v) {
    float r;
    // gfx1250 hardware transcendental tanh; v_nop covers the TRANS RAW hazard
    asm volatile("v_tanh_f32 %0, %1\n\tv_nop" : "=v"(r) : "v"(v));
    return r;
}

// ---------------------------------------------------------------- K1: tables
__global__ __launch_bounds__(128) void k1_prep(
    const float* __restrict__ fc1_w, const float* __restrict__ fc1_b,
    const float* __restrict__ fc2_w, const float* __restrict__ fc2_b,
    const float* __restrict__ proj_w, const float* __restrict__ proj_b,
    float* __restrict__ ws)
{
    const int tid = threadIdx.x;
    // Gpad[64][52]
    for (int i = tid; i < 64 * 52; i += 128) {
        int p = i / 52, t = i - p * 52;
        float v;
        if (t == 50) v = 1.0f;                    // mean pass-through column
        else if (t == 51 || p >= 60) v = 0.0f;    // padding
        else {
            float s = 0.0f;
            for (int j = 0; j < 32; ++j) s += fc2_w[p * 32 + j] * fc1_w[j * 50 + t];
            v = s;
        }
        ws[i] = v;
    }
    // outbias[64][12]
    for (int i = tid; i < 64 * 12; i += 128) {
        int p = i / 12, w = i - p * 12;
        float v = 0.0f;
        if (p < 60) {
            float t2 = fc2_b[p];
            for (int j = 0; j < 32; ++j) t2 += fc2_w[p * 32 + j] * fc1_b[j];
            float ps = 0.0f;
            for (int r = 0; r < 25; ++r) ps += proj_w[w * 25 + r];
            v = proj_b[w] + ps * t2;
        }
        ws[3328 + i] = v;
    }
}

// ------------------------------------------------------- K2: mean + ESN + q
__global__ __launch_bounds__(256) void k2_recurrence(
    const float* __restrict__ x, const float* __restrict__ w_in,
    const float* __restrict__ w_res, const float* __restrict__ proj_w,
    float* __restrict__ out)
{
    __shared__ __align__(16) float s_winT[12][28];   // [w][r]
    __shared__ __align__(16) float s_wresT[25][28];  // [j][r]
    __shared__ __align__(16) float s_projT[25][12];  // [r][w]
    __shared__ __align__(16) float s_wsum[32];       // rowsum(w_in)

    const int tid = threadIdx.x;
    for (int i = tid; i < 12 * 25; i += 256) { int w = i / 25, r = i - w * 25; s_winT[w][r]  = w_in[r * 12 + w]; }
    for (int i = tid; i < 25 * 25; i += 256) { int j = i / 25, r = i - j * 25; s_wresT[j][r] = w_res[r * 25 + j]; }
    for (int i = tid; i < 25 * 12; i += 256) { int r = i / 12, w = i - r * 12; s_projT[r][w] = proj_w[w * 25 + r]; }
    if (tid < 25) {
        float s = 0.0f;
        for (int w = 0; w < 12; ++w) s += w_in[tid * 12 + w];
        s_wsum[tid] = s;
    }
    __syncthreads();

    const int n = blockIdx.x * 256 + tid;           // exact: 431*256 = 110336
    const int b = n / NC;
    const int c = n - b * NC;
    const float* xb = x + (size_t)b * SROW + c;     // x[b, t, c] = xb[t*862]
    float* ob = out + (size_t)b * SROW + c;         // out[b, s, c] = ob[s*862]

    // mean over time (lane-coalesced over c)
    float acc = 0.0f;
#pragma unroll 8
    for (int t = 0; t < 720; ++t) acc += xb[(size_t)t * NC];
    const float mean = acc * (1.0f / 720.0f);
    // stage mean into the K=50 GEMM column (rows 600..611)
#pragma unroll
    for (int w = 0; w < 12; ++w) ob[(size_t)(600 + w) * NC] = mean;

    // mean-centering folded in: -mean * rowsum(W_in), hoisted out of the t loop
    float negms[25];
#pragma unroll
    for (int r = 0; r < 25; ++r) negms[r] = -mean * s_wsum[r];

    float h[25];
#pragma unroll
    for (int r = 0; r < 25; ++r) h[r] = 0.0f;

    for (int t = 0; t < 60; ++t) {
        float pre[25];
        // w = 0 input contribution initializes pre (folds the bias term into an FMA)
        {
            float xv = xb[(size_t)t * NC];
#pragma unroll
            for (int r = 0; r < 25; ++r) pre[r] = fmaf(s_winT[0][r], xv, negms[r]);
        }
#pragma unroll
        for (int w = 1; w < 12; ++w) {
            float xv = xb[(size_t)(w * 60 + t) * NC];
#pragma unroll
            for (int r = 0; r < 25; ++r) pre[r] += s_winT[w][r] * xv;
        }
        // recurrent contribution
#pragma unroll
        for (int j = 0; j < 25; ++j) {
            float hj = h[j];
#pragma unroll
            for (int r = 0; r < 25; ++r) pre[r] += s_wresT[j][r] * hj;
        }
#pragma unroll
        for (int r = 0; r < 25; ++r) h[r] = tanh_hw(pre[r]);

        if (t >= 10) {
            float q[12];
#pragma unroll
            for (int w = 0; w < 12; ++w) q[w] = 0.0f;
#pragma unroll
            for (int r = 0; r < 25; ++r) {
                float hr = h[r];
#pragma unroll
                for (int w = 0; w < 12; ++w) q[w] += s_projT[r][w] * hr;
            }
            const int row = (t - 10) * 12;
#pragma unroll
            for (int w = 0; w < 12; ++w) ob[(size_t)(row + w) * NC] = q[w];
        }
    }
}

// --------------------------------------------------- K3: WMMA head GEMM
// One wave per block, full EXEC for all WMMAs. block = (w, group of 8 n-tiles).
__global__ __launch_bounds__(32) void k3_head(
    const float* __restrict__ ws, float* __restrict__ out)
{
    const int lane = threadIdx.x;
    const int half = lane >> 4;
    const int lm   = lane & 15;
    const int w    = blockIdx.x % 12;
    const int g    = blockIdx.x / 12;   // 0..861

    // A fragments: Gpad[p, t]; A layout: M=lane%16, K=reg + 2*(lane/16)
    v2f a[4][13];
#pragma unroll
    for (int mt = 0; mt < 4; ++mt) {
        const float* gp = ws + (size_t)(16 * mt + lm) * 52 + 2 * half;
#pragma unroll
        for (int kt = 0; kt < 13; ++kt) {
            a[mt][kt].x = gp[4 * kt + 0];
            a[mt][kt].y = gp[4 * kt + 1];
        }
    }
    // Per-lane output bias, hoisted out of the n-tile loop (table is 768 entries,
    // p <= 63 -> index <= 767 stays in bounds; p >= 60 rows are never stored).
    const float* obias = ws + 3328;
    float obv[4][8];
#pragma unroll
    for (int mt = 0; mt < 4; ++mt)
#pragma unroll
        for (int v = 0; v < 8; ++v) {
            const int p = 16 * mt + v + 8 * half;
            obv[mt][v] = obias[p * 12 + w];
        }

#pragma unroll 1
    for (int i = 0; i < 8; ++i) {
        const int n = (g * 8 + i) * 16 + lm;        // exact: 862*8*16 = 110336
        const int b = n / NC;
        const int c = n - b * NC;
        float* base = out + (size_t)b * SROW + (size_t)w * NC + c;  // row p=0 / t'=0
        const float* brd = base + (size_t)(2 * half) * PSTEP;

        // Batch-load all 26 B fragments first: one deep clause, 13x the MLP of a
        // per-k-step load/wait pattern.
        float bb0[13], bb1[13];
#pragma unroll
        for (int kt = 0; kt < 13; ++kt) {
            bb0[kt] = brd[(size_t)(4 * kt + 0) * PSTEP];
            bb1[kt] = brd[(size_t)(4 * kt + 1) * PSTEP];
        }
        if (half == 1) bb1[12] = 0.0f;              // t'=51 zero column

        v8f c0 = {}, c1 = {}, c2 = {}, c3 = {};
#pragma unroll
        for (int kt = 0; kt < 13; ++kt) {
            v2f bf; bf.x = bb0[kt]; bf.y = bb1[kt];
            c0 = __builtin_amdgcn_wmma_f32_16x16x4_f32(false, a[0][kt], false, bf, (short)0, c0, false, false);
            c1 = __builtin_amdgcn_wmma_f32_16x16x4_f32(false, a[1][kt], false, bf, (short)0, c1, false, false);
            c2 = __builtin_amdgcn_wmma_f32_16x16x4_f32(false, a[2][kt], false, bf, (short)0, c2, false, false);
            c3 = __builtin_amdgcn_wmma_f32_16x16x4_f32(false, a[3][kt], false, bf, (short)0, c3, false, false);
        }
        // Epilogue: D layout row = v + 8*half (+16*mt), col = lane%16
#pragma unroll
        for (int mt = 0; mt < 4; ++mt) {
            v8f cc = (mt == 0) ? c0 : (mt == 1) ? c1 : (mt == 2) ? c2 : c3;
#pragma unroll
            for (int v = 0; v < 8; ++v) {
                const int p = 16 * mt + v + 8 * half;
                if (p < 60) {
                    base[(size_t)p * PSTEP] = cc[v] + obv[mt][v];
                }
            }
        }
    }
}

extern "C" void kernel_launch(void* const* d_in, const int* in_sizes, int n_in,
                              void* d_out, int out_size, void* d_ws, size_t ws_size,
                              hipStream_t stream) {
    (void)in_sizes; (void)n_in; (void)out_size; (void)ws_size;
    const float* x      = (const float*)d_in[0];
    const float* w_in   = (const float*)d_in[1];
    const float* w_res  = (const float*)d_in[2];
    const float* fc1_w  = (const float*)d_in[3];
    const float* fc1_b  = (const float*)d_in[4];
    const float* fc2_w  = (const float*)d_in[5];
    const float* fc2_b  = (const float*)d_in[6];
    const float* proj_w = (const float*)d_in[7];
    const float* proj_b = (const float*)d_in[8];
    float* out = (float*)d_out;
    float* ws  = (float*)d_ws;   // needs 16 KiB

    k1_prep<<<dim3(1), dim3(128), 0, stream>>>(fc1_w, fc1_b, fc2_w, fc2_b, proj_w, proj_b, ws);
    k2_recurrence<<<dim3(431), dim3(256), 0, stream>>>(x, w_in, w_res, proj_w, out);
    k3_head<<<dim3(10344), dim3(32), 0, stream>>>(ws, out);
}